// AI4DEM_80358838108610
// MI455X (gfx1250) — compile-verified
//
#include <hip/hip_runtime.h>
#include <cmath>
#include <cstdint>

// ---------------- problem geometry (fixed by the reference) ----------------
namespace {
constexpr int H_ = 128;           // z extent
constexpr int W_ = 256;           // y and x extent
constexpr int N1 = H_ * W_ * W_;  // 8,388,608 cells per field

constexpr int TX = 32, TY = 4, TZ = 4;              // interior tile per block
constexpr int HX = TX + 4, HY = TY + 4, HZ = TZ + 4; // halo tile 36 x 8 x 8
constexpr int NH = HX * HY * HZ;                     // 2304 halo cells
constexpr int NT = TX * TY * TZ;                     // 512 threads / block

constexpr float KN     = 600000.0f;
constexpr float EPS_   = 1e-4f;
constexpr float R_P    = 0.05f;
constexpr float TWO_RP = 0.1f;   // 2*R_P == 2*D
constexpr float WEXT   = 12.8f;  // 256 * 0.05
}

struct alignas(16) F4 { float x, y, z, w; };
struct alignas(8)  F2 { float x, y; };

typedef float v2f __attribute__((ext_vector_type(2)));
__device__ __forceinline__ v2f splat2(float s) { return (v2f){s, s}; }

// ---------------- CDNA5 async global->LDS gather (probe-verified) ----------
#if defined(__HIP_DEVICE_COMPILE__) && \
    __has_builtin(__builtin_amdgcn_global_load_async_to_lds_b32)
#define HAVE_ASYNC_G2L 1
#else
#define HAVE_ASYNC_G2L 0
#endif

#if HAVE_ASYNC_G2L
typedef __attribute__((address_space(1))) int g_int;   // global
typedef __attribute__((address_space(3))) int l_int;   // LDS

__device__ __forceinline__ void g2l_b32(const float* gp, float* lp) {
  __builtin_amdgcn_global_load_async_to_lds_b32(
      (g_int*)(float*)gp,
      (l_int*)lp,
      /*offset=*/0, /*cpol=*/0);
}
#endif

__device__ __forceinline__ void wait_async_then_barrier() {
#if HAVE_ASYNC_G2L
#if __has_builtin(__builtin_amdgcn_s_wait_asynccnt)
  __builtin_amdgcn_s_wait_asynccnt(0);
#else
  asm volatile("s_wait_asynccnt 0" ::: "memory");
#endif
#endif
  __syncthreads();
}

// ---------------------------- main kernel ----------------------------------
__global__ __launch_bounds__(NT)
void AI4DEM_forces_kernel(const float* __restrict__ px, const float* __restrict__ py,
                          const float* __restrict__ pz, const float* __restrict__ pvx,
                          const float* __restrict__ pvy, const float* __restrict__ pvz,
                          const float* __restrict__ pm, float* __restrict__ out,
                          float eta, float eta_wall) {
  static_assert(NH * (int)sizeof(F4) + NH * (int)sizeof(F2) <= 65536, "LDS budget");

  __shared__ F4 s_a[NH];  // x, y, z, vx
  __shared__ F2 s_b[NH];  // vy, vz

  const int x0 = blockIdx.x * TX;
  const int y0 = blockIdx.y * TY;
  const int z0 = blockIdx.z * TZ;
  const int tx = (int)threadIdx.x, ty = (int)threadIdx.y, tz = (int)threadIdx.z;
  const int tid = (tz * TY + ty) * TX + tx;

  // ---- stage halo tile (periodic wrap) into LDS via async gather ----
  for (int c = tid; c < NH; c += NT) {
    const int lx = c % HX;
    const int t  = c / HX;
    const int ly = t & (HY - 1);
    const int lz = t >> 3;
    const int gx = (x0 + lx - 2) & (W_ - 1);
    const int gy = (y0 + ly - 2) & (W_ - 1);
    const int gz = (z0 + lz - 2) & (H_ - 1);
    const int g  = (gz << 16) | (gy << 8) | gx;
#if HAVE_ASYNC_G2L
    g2l_b32(px  + g, &s_a[c].x);
    g2l_b32(py  + g, &s_a[c].y);
    g2l_b32(pz  + g, &s_a[c].z);
    g2l_b32(pvx + g, &s_a[c].w);
    g2l_b32(pvy + g, &s_b[c].x);
    g2l_b32(pvz + g, &s_b[c].y);
#else
    s_a[c] = F4{px[g], py[g], pz[g], pvx[g]};
    s_b[c] = F2{pvy[g], pvz[g]};
#endif
  }
  wait_async_then_barrier();

  // ---- 125-point stencil from LDS, packed 2-wide for V_PK_*_F32 ----
  const int c0 = ((tz + 2) * HY + (ty + 2)) * HX + (tx + 2);
  const F4 ca = s_a[c0];
  const F2 cb = s_b[c0];
  const float cx = ca.x, cy = ca.y, cz = ca.z;
  const float cvx = ca.w, cvy = cb.x, cvz = cb.y;

  v2f fx2 = splat2(0.0f), fy2 = splat2(0.0f), fz2 = splat2(0.0f);
  float fxs = 0.0f, fys = 0.0f, fzs = 0.0f;

  // packed pair of stencil taps (two neighbor cells per call)
  auto pair_term = [&](int n0, int n1) {
    const F4 a0 = s_a[n0]; const F4 a1 = s_a[n1];
    const F2 b0 = s_b[n0]; const F2 b1 = s_b[n1];
    v2f rx = {cx - a0.x, cx - a1.x};
    v2f ry = {cy - a0.y, cy - a1.y};
    v2f rz = {cz - a0.z, cz - a1.z};
    v2f sq = rx * rx + ry * ry + rz * rz;
    v2f dist, denom, rin;
    dist.x  = __builtin_amdgcn_sqrtf(sq.x);
    dist.y  = __builtin_amdgcn_sqrtf(sq.y);
    denom.x = fmaxf(EPS_, dist.x);
    denom.y = fmaxf(EPS_, dist.y);
    rin.x   = __builtin_amdgcn_rcpf(denom.x);
    rin.y   = __builtin_amdgcn_rcpf(denom.y);
    v2f dvx = {cvx - a0.w, cvx - a1.w};
    v2f dvy = {cvy - b0.x, cvy - b1.x};
    v2f dvz = {cvz - b0.y, cvz - b1.y};
    v2f dot = dvx * rx + dvy * ry + dvz * rz;
    v2f coef = (splat2(KN) * (dist - splat2(TWO_RP)) + splat2(eta) * (dot * rin)) * rin;
    coef.x = (dist.x < TWO_RP) ? coef.x : 0.0f;
    coef.y = (dist.y < TWO_RP) ? coef.y : 0.0f;
    fx2 += coef * rx;
    fy2 += coef * ry;
    fz2 += coef * rz;
  };

  // scalar tap (self tap at offset 0 contributes exactly +0.0, same as reference)
  auto single_term = [&](int n) {
    const F4 na = s_a[n];
    const F2 nb = s_b[n];
    const float rx = cx - na.x;
    const float ry = cy - na.y;
    const float rz = cz - na.z;
    const float sq = rx * rx + ry * ry + rz * rz;
    const float dist  = __builtin_amdgcn_sqrtf(sq);
    const float denom = fmaxf(EPS_, dist);
    const float rin   = __builtin_amdgcn_rcpf(denom);
    const float dvx = cvx - na.w;
    const float dvy = cvy - nb.x;
    const float dvz = cvz - nb.y;
    const float dot = dvx * rx + dvy * ry + dvz * rz;
    float coef = (KN * (dist - TWO_RP) + eta * (dot * rin)) * rin;
    coef = (dist < TWO_RP) ? coef : 0.0f;
    fxs += coef * rx;
    fys += coef * ry;
    fzs += coef * rz;
  };

#pragma unroll 1
  for (int dz = -2; dz <= 2; ++dz) {
    const int pbase = c0 + dz * (HX * HY);
#pragma unroll
    for (int dy = -2; dy <= 2; ++dy) {
      const int base = pbase + dy * HX;
      pair_term(base - 2, base - 1);   // dx = -2, -1
      pair_term(base + 1, base + 2);   // dx = +1, +2
    }
    // dx == 0 column of this plane, paired across dy:
    pair_term(pbase - 2 * HX, pbase - HX);   // (dy=-2, dy=-1)
    pair_term(pbase + HX, pbase + 2 * HX);   // (dy=+1, dy=+2)
    single_term(pbase);                      // (dy=0, dx=0); dz==0 -> self tap = exact 0
  }

  float fx = fxs + fx2.x + fx2.y;
  float fy = fys + fy2.x + fy2.y;
  float fz = fzs + fz2.x + fz2.y;

  // ---- wall forces (center cell only) ----
  const int g = ((z0 + tz) << 16) | ((y0 + ty) << 8) | (x0 + tx);
  const float m = pm[g];

  const float il  = (cx != 0.0f && cx < R_P) ? 1.0f : 0.0f;
  const float ir  = (cx > (WEXT - TWO_RP))   ? 1.0f : 0.0f;
  const float ib  = (cy != 0.0f && cy < R_P) ? 1.0f : 0.0f;
  const float it  = (cy > (WEXT - TWO_RP))   ? 1.0f : 0.0f;
  const float ifw = (cz != 0.0f && cz < R_P) ? 1.0f : 0.0f;
  const float ibk = (cz > (WEXT - TWO_RP))   ? 1.0f : 0.0f;

  fx += KN * il  * m * (R_P - cx) - KN * ir  * m * ((cx - WEXT) + TWO_RP)
        - eta_wall * cvx * (il + ir) * m;
  fy += KN * ib  * m * (R_P - cy) - KN * it  * m * ((cy - WEXT) + TWO_RP)
        - eta_wall * cvy * (ib + it) * m;
  fz += KN * ifw * m * (R_P - cz) - KN * ibk * m * ((cz - WEXT) + TWO_RP)
        - eta_wall * cvz * (ifw + ibk) * m;

  out[g]          = fx;
  out[N1 + g]     = fy;
  out[2 * N1 + g] = fz;
}

// ---------------------------- host launcher --------------------------------
extern "C" void kernel_launch(void* const* d_in, const int* in_sizes, int n_in,
                              void* d_out, int out_size, void* d_ws, size_t ws_size,
                              hipStream_t stream) {
  (void)in_sizes; (void)n_in; (void)out_size; (void)d_ws; (void)ws_size;

  const float* px  = (const float*)d_in[0];
  const float* py  = (const float*)d_in[1];
  const float* pz  = (const float*)d_in[2];
  const float* pvx = (const float*)d_in[3];
  const float* pvy = (const float*)d_in[4];
  const float* pvz = (const float*)d_in[5];
  const float* pm  = (const float*)d_in[6];
  float* out = (float*)d_out;

  // Exact constant derivation from the reference (double precision on host).
  const double pi_exact = 3.141592653589793;
  const double alpha = 0.6931471805599453 / pi_exact;       // -log(0.5)/pi
  const double gamma = alpha / std::sqrt(alpha * alpha + 1.0);
  const double mass  = 4.0 / 3.0 * 3.1415 * (0.05 * 0.05 * 0.05) * 2700.0;
  const float eta      = (float)(2.0 * gamma * std::sqrt(600000.0 * mass / 2.0));
  const float eta_wall = (float)(2.0 * gamma * std::sqrt(600000.0 * mass));

  dim3 grid(W_ / TX, W_ / TY, H_ / TZ);   // 8 x 64 x 32 blocks
  dim3 block(TX, TY, TZ);                 // 512 threads (16 wave32)
  AI4DEM_forces_kernel<<<grid, block, 0, stream>>>(px, py, pz, pvx, pvy, pvz, pm,
                                                   out, eta, eta_wall);
}